// P2MLoss_73933567033560
// MI455X (gfx1250) — compile-verified
//
#include <hip/hip_runtime.h>
#include <hip/hip_bf16.h>
#include <math.h>

// ---------------------------------------------------------------------------
// P2M loss on gfx1250 (MI455X, wave32, WMMA).
// Chamfer pairwise distances via V_WMMA_F32_16X16X4_F32:
//   A row m = (-2x0, -2x1, -2x2, |x|^2), B col n = (y0, y1, y2, 1)
//   D[m][n] = -2 x.y + |x|^2 ; add |y|^2 (staged in LDS) per column.
// gt points staged once per block in LDS (128KB) + LDS column-min (32KB);
// 8 waves/block share the staged copy; ds_min_u32 merges column minima.
// ---------------------------------------------------------------------------

typedef __attribute__((ext_vector_type(2))) float v2f;
typedef __attribute__((ext_vector_type(8))) float v8f;

// Order-preserving float <-> uint encoding (for atomicMin on float values).
__device__ __forceinline__ unsigned fenc_ordered(float f) {
  unsigned u = __float_as_uint(f);
  return (u & 0x80000000u) ? ~u : (u | 0x80000000u);
}
__device__ __forceinline__ float fdec_ordered(unsigned u) {
  unsigned b = (u & 0x80000000u) ? (u & 0x7FFFFFFFu) : ~u;
  return __uint_as_float(b);
}

#define ACC_ROW   0   // [0..2]  sum over (b,v) of min_m d
#define ACC_COL   3   // [3..5]  sum over (b,m) of min_v d
#define ACC_EDGE  6   // [6..8]
#define ACC_NORM  9   // [9..11]
#define ACC_LAP   12  // [12..14]
#define ACC_MOVE  15  // [15..17]
#define ACC_SLOTS 32

__global__ __launch_bounds__(32) void p2m_zero_acc(float* acc) {
  if (threadIdx.x < ACC_SLOTS) acc[threadIdx.x] = 0.0f;
}

__global__ void p2m_init_colmin(unsigned* colmin, int n) {
  int i = blockIdx.x * blockDim.x + threadIdx.x;
  if (i < n) colmin[i] = 0xFFFFFFFFu;  // above enc(+inf)
}

// ---------------------------------------------------------------------------
// Chamfer: 256-thread block (8 wave32), one 16-row pred tile per wave
// (block covers 128 rows). Dynamic LDS: float4 gt cache [M] + u32 colmin [M].
// C/D layout: vgpr j, lane l -> row = tile*16 + j + 8*(l>>4), col = (l&15).
// ---------------------------------------------------------------------------
__global__ __launch_bounds__(256)
void p2m_chamfer_wmma(const float* __restrict__ pred,  // [B,V,3]
                      const float* __restrict__ gt,    // [B,M,3]
                      int V, int M,
                      unsigned* __restrict__ colmin,   // [B*M] ordered-enc
                      int* __restrict__ knn,           // [B*V]
                      float* __restrict__ rowsum_acc) {
  extern __shared__ unsigned char smem_raw[];
  float4*   s_gt = (float4*)smem_raw;                         // M * 16 B
  unsigned* s_cm = (unsigned*)(smem_raw + (size_t)M * 16);    // M * 4 B

  const int tid  = threadIdx.x;
  const int wave = tid >> 5;
  const int lane = tid & 31;
  const int half = lane >> 4;   // selects K pair (0,1) vs (2,3)
  const int l16  = lane & 15;
  const int b    = blockIdx.y;

  // ---- Stage gt into LDS as (x, y, z, |y|^2); init LDS column minima ----
  const float* gbase = gt + (size_t)b * M * 3;
  for (int m = tid; m < M; m += 256) {
    const float* g = gbase + (size_t)m * 3;
    float gx = g[0], gy = g[1], gz = g[2];
    s_gt[m] = make_float4(gx, gy, gz, gx * gx + gy * gy + gz * gz);
    s_cm[m] = 0xFFFFFFFFu;
  }
  __syncthreads();

  // ---- A fragment (constant for the whole M loop) ----
  const int tile = blockIdx.x * 8 + wave;
  const int vrow = tile * 16 + l16;
  float px = 0.f, py = 0.f, pz = 0.f;
  if (vrow < V) {
    const float* p = pred + ((size_t)b * V + vrow) * 3;
    px = p[0]; py = p[1]; pz = p[2];
  }
  float xn2 = px * px + py * py + pz * pz;
  v2f a;
  if (half == 0) { a.x = -2.f * px; a.y = -2.f * py; }
  else           { a.x = -2.f * pz; a.y = xn2;       }

  // Row validity per C vgpr j (uniform within each 16-lane half).
  bool rvalid[8];
#pragma unroll
  for (int j = 0; j < 8; ++j) rvalid[j] = (tile * 16 + j + 8 * half) < V;

  // Running per-lane row minima (this lane's column slice of M).
  float rmin[8];
  int   rarg[8];
#pragma unroll
  for (int j = 0; j < 8; ++j) { rmin[j] = INFINITY; rarg[j] = 0; }

  // ---- Main loop: 2 WMMA tiles per iteration (M % 32 == 0) ----
  for (int mt = 0; mt < M; mt += 32) {
    const int col0 = mt + l16;
    const int col1 = col0 + 16;
    float4 g0 = s_gt[col0];
    float4 g1 = s_gt[col1];
    v2f b0, b1;
    if (half == 0) { b0.x = g0.x; b0.y = g0.y; b1.x = g1.x; b1.y = g1.y; }
    else           { b0.x = g0.z; b0.y = 1.f;  b1.x = g1.z; b1.y = 1.f;  }

    v8f c0 = {}, c1 = {};
    c0 = __builtin_amdgcn_wmma_f32_16x16x4_f32(false, a, false, b0,
                                               (short)0, c0, false, false);
    c1 = __builtin_amdgcn_wmma_f32_16x16x4_f32(false, a, false, b1,
                                               (short)0, c1, false, false);

    float cm0 = INFINITY, cm1 = INFINITY;
#pragma unroll
    for (int j = 0; j < 8; ++j) {
      float d0 = c0[j] + g0.w;
      float d1 = c1[j] + g1.w;
      if (d0 < rmin[j]) { rmin[j] = d0; rarg[j] = col0; }
      if (d1 < rmin[j]) { rmin[j] = d1; rarg[j] = col1; }
      cm0 = fminf(cm0, rvalid[j] ? d0 : INFINITY);
      cm1 = fminf(cm1, rvalid[j] ? d1 : INFINITY);
    }
    // LDS atomic min merges both 16-lane halves (same address) and all
    // 8 waves of the block -- no shuffles, no divergence in the loop.
    atomicMin(&s_cm[col0], fenc_ordered(cm0));
    atomicMin(&s_cm[col1], fenc_ordered(cm1));
  }

  // ---- Row min+argmin: cross-lane butterfly once, at the end ----
#pragma unroll
  for (int j = 0; j < 8; ++j) {
    float v = rmin[j];
    int   ai = rarg[j];
#pragma unroll
    for (int m = 1; m < 16; m <<= 1) {
      float ov = __shfl_xor(v, m, 32);
      int   oa = __shfl_xor(ai, m, 32);
      if (ov < v) { v = ov; ai = oa; }
    }
    rmin[j] = v; rarg[j] = ai;
  }
  if (l16 == 0) {  // lanes 0 / 16 own rows [tile*16..+7] / [+8..+15]
    float s = 0.f;
#pragma unroll
    for (int j = 0; j < 8; ++j) {
      int rrow = tile * 16 + j + 8 * half;
      if (rrow < V) {
        knn[(size_t)b * V + rrow] = rarg[j];
        s += rmin[j];
      }
    }
    atomicAdd(rowsum_acc, s);
  }

  // ---- Merge block-local column minima into global (one atomic/col/block) --
  __syncthreads();
  for (int m = tid; m < M; m += 256) {
    atomicMin(&colmin[(size_t)b * M + m], s_cm[m]);
  }
}

// Sum decoded column minima -> chamfer axis-1 term.
__global__ void p2m_colsum(const unsigned* __restrict__ colmin, int n,
                           float* __restrict__ acc) {
  int i = blockIdx.x * blockDim.x + threadIdx.x;
  float v = (i < n) ? fdec_ordered(colmin[i]) : 0.f;
#pragma unroll
  for (int m = 16; m >= 1; m >>= 1) v += __shfl_xor(v, m, 32);
  if ((threadIdx.x & 31) == 0) atomicAdd(acc, v);
}

// Edge + normal losses. Reference quirk: sel = knn.reshape(-1)[e0] ->
// always batch-0 knn.
__global__ void p2m_edge_normal(const float* __restrict__ pred,
                                const float* __restrict__ gtn,  // [B,M,3]
                                const int* __restrict__ edges,  // [E,2]
                                const int* __restrict__ knn,    // [B*V]
                                int V, int E, int M, int B,
                                float* __restrict__ edge_acc,
                                float* __restrict__ normal_acc) {
  int i = blockIdx.x * blockDim.x + threadIdx.x;
  float es = 0.f, ns = 0.f;
  if (i < B * E) {
    int b = i / E, e = i - b * E;
    int e0 = edges[2 * e], e1 = edges[2 * e + 1];
    const float* p0 = pred + ((size_t)b * V + e0) * 3;
    const float* p1 = pred + ((size_t)b * V + e1) * 3;
    float dx = p0[0] - p1[0], dy = p0[1] - p1[1], dz = p0[2] - p1[2];
    es = dx * dx + dy * dy + dz * dz;
    float inv = 1.0f / fmaxf(sqrtf(es), 1e-12f);
    int sel = knn[e0];                          // batch-0 knn (reshape quirk)
    const float* gn = gtn + ((size_t)b * M + sel) * 3;
    float nx = gn[0], ny = gn[1], nz = gn[2];
    float ninv = 1.0f / fmaxf(sqrtf(nx * nx + ny * ny + nz * nz), 1e-12f);
    ns = fabsf((dx * nx + dy * ny + dz * nz) * inv * ninv);
  }
#pragma unroll
  for (int m = 16; m >= 1; m >>= 1) {
    es += __shfl_xor(es, m, 32);
    ns += __shfl_xor(ns, m, 32);
  }
  if ((threadIdx.x & 31) == 0) {
    atomicAdd(edge_acc, es);
    atomicAdd(normal_acc, ns);
  }
}

// Laplacian + move losses.
__global__ void p2m_lap_move(const float* __restrict__ pred,
                             const float* __restrict__ before,
                             const int* __restrict__ lapidx,  // [V,10]
                             int V, int B,
                             float* __restrict__ lap_acc,
                             float* __restrict__ move_acc) {
  int i = blockIdx.x * blockDim.x + threadIdx.x;
  float ls = 0.f, ms = 0.f;
  if (i < B * V) {
    int b = i / V, v = i - b * V;
    const int* li = lapidx + (size_t)v * 10;
    float inv = 1.0f / (float)li[9];
    float sbx = 0, sby = 0, sbz = 0, spx = 0, spy = 0, spz = 0;
#pragma unroll
    for (int k = 0; k < 8; ++k) {
      int idx = li[k];
      if (idx >= 0) {
        const float* pb = before + ((size_t)b * V + idx) * 3;
        const float* pp = pred   + ((size_t)b * V + idx) * 3;
        sbx += pb[0]; sby += pb[1]; sbz += pb[2];
        spx += pp[0]; spy += pp[1]; spz += pp[2];
      }
    }
    const float* b0 = before + ((size_t)b * V + v) * 3;
    const float* p0 = pred   + ((size_t)b * V + v) * 3;
    float dx = (b0[0] - sbx * inv) - (p0[0] - spx * inv);
    float dy = (b0[1] - sby * inv) - (p0[1] - spy * inv);
    float dz = (b0[2] - sbz * inv) - (p0[2] - spz * inv);
    ls = dx * dx + dy * dy + dz * dz;
    float mx = b0[0] - p0[0], my = b0[1] - p0[1], mz = b0[2] - p0[2];
    ms = mx * mx + my * my + mz * mz;
  }
#pragma unroll
  for (int m = 16; m >= 1; m >>= 1) {
    ls += __shfl_xor(ls, m, 32);
    ms += __shfl_xor(ms, m, 32);
  }
  if ((threadIdx.x & 31) == 0) {
    atomicAdd(lap_acc, ls);
    atomicAdd(move_acc, ms);
  }
}

__global__ void p2m_finalize(const float* __restrict__ acc,
                             float* __restrict__ out, int B, int M,
                             int V0, int V1, int V2,
                             int E0, int E1, int E2) {
  if (threadIdx.x == 0 && blockIdx.x == 0) {
    const float lapc[3] = {0.2f, 1.0f, 1.0f};
    int Vs[3] = {V0, V1, V2};
    int Es[3] = {E0, E1, E2};
    float chamfer = 0.f, lap = 0.f, move = 0.f, edge = 0.f, nor = 0.f;
    for (int i = 0; i < 3; ++i) {
      float invBV = 1.0f / (float)(B * Vs[i]);
      float invBE = 1.0f / (float)(B * Es[i]);
      chamfer += acc[ACC_COL + i] / (float)(B * M) + acc[ACC_ROW + i] * invBV;
      edge    += acc[ACC_EDGE + i] * invBE;            // 3*mean == sum/(B*E)
      nor     += acc[ACC_NORM + i] * invBE;
      lap     += lapc[i] * acc[ACC_LAP + i] * invBV;   // LAPC*3*mean == LAPC*sum/(B*V)
      if (i > 0) move += lapc[i] * acc[ACC_MOVE + i] * invBV;
    }
    out[0] = chamfer + 0.5f * lap + 0.1f * move + 0.1f * edge + 0.00016f * nor;
  }
}

extern "C" void kernel_launch(void* const* d_in, const int* in_sizes, int n_in,
                              void* d_out, int out_size, void* d_ws, size_t ws_size,
                              hipStream_t stream) {
  const int B = 2, M = 8192;
  const int Vs[3] = {642, 2562, 10242};
  const int Es[3] = {1920, 7680, 30720};

  // setup_inputs() dict order:
  // 0 gt_coord, 1 gt_normal, then per level: pred_coord, pred_before, edges, laplace_idx
  const float* gt  = (const float*)d_in[0];
  const float* gtn = (const float*)d_in[1];
  const float* pred[3]   = {(const float*)d_in[2], (const float*)d_in[6],  (const float*)d_in[10]};
  const float* before[3] = {(const float*)d_in[3], (const float*)d_in[7],  (const float*)d_in[11]};
  const int*   edges[3]  = {(const int*)d_in[4],   (const int*)d_in[8],    (const int*)d_in[12]};
  const int*   lapidx[3] = {(const int*)d_in[5],   (const int*)d_in[9],    (const int*)d_in[13]};

  // Workspace: [0,256) accumulators; [256,+84K) knn (B*Vmax, reused);
  // then colmin (B*M u32, reused per level).
  float*    acc    = (float*)d_ws;
  int*      knn    = (int*)((char*)d_ws + 256);
  unsigned* colmin = (unsigned*)((char*)d_ws + 256 + 84 * 1024);

  const size_t lds_bytes = (size_t)M * 16 + (size_t)M * 4;  // 160 KB (<320 KB)

  p2m_zero_acc<<<1, 32, 0, stream>>>(acc);

  for (int i = 0; i < 3; ++i) {
    const int V = Vs[i], E = Es[i];
    const int nCol = B * M;
    p2m_init_colmin<<<(nCol + 255) / 256, 256, 0, stream>>>(colmin, nCol);

    dim3 grid((V + 127) / 128, B);  // 8 row tiles (128 rows) per block
    p2m_chamfer_wmma<<<grid, 256, lds_bytes, stream>>>(pred[i], gt, V, M,
                                                       colmin, knn,
                                                       acc + ACC_ROW + i);
    p2m_colsum<<<(nCol + 255) / 256, 256, 0, stream>>>(colmin, nCol,
                                                       acc + ACC_COL + i);
    const int nBE = B * E;
    p2m_edge_normal<<<(nBE + 255) / 256, 256, 0, stream>>>(
        pred[i], gtn, edges[i], knn, V, E, M, B,
        acc + ACC_EDGE + i, acc + ACC_NORM + i);
    const int nBV = B * V;
    p2m_lap_move<<<(nBV + 255) / 256, 256, 0, stream>>>(
        pred[i], before[i], lapidx[i], V, B,
        acc + ACC_LAP + i, acc + ACC_MOVE + i);
  }

  p2m_finalize<<<1, 32, 0, stream>>>(acc, (float*)d_out, B, M,
                                     Vs[0], Vs[1], Vs[2], Es[0], Es[1], Es[2]);
}